// SAGPool_56882546868357
// MI455X (gfx1250) — compile-verified
//
#include <hip/hip_runtime.h>
#include <hip/hip_bf16.h>
#include <math.h>

// ---- static problem sizes (match reference) ----
#define BGR   1024
#define NPER  128
#define FDIM  256
#define NTOT  (BGR * NPER)      // 131072
#define ETOT  (NTOT * 32)       // 4194304
#define KEEP  103               // ceil(0.8 * 128)
#define BK    (BGR * KEEP)      // 105472

typedef __attribute__((ext_vector_type(2))) float v2f;
typedef __attribute__((ext_vector_type(4))) float v4f;
typedef __attribute__((ext_vector_type(8))) float v8f;

// -------- 1. init: deg = 1 (self loop), agg = 0, new_idx = -1 --------
__global__ __launch_bounds__(256) void init_kernel(float* __restrict__ deg,
                                                   float* __restrict__ agg,
                                                   int* __restrict__ new_idx) {
    int i = blockIdx.x * 256 + threadIdx.x;
    if (i < NTOT) { deg[i] = 1.0f; agg[i] = 0.0f; new_idx[i] = -1; }
}

// -------- 2. h = x @ W via V_WMMA_F32_16X16X4_F32 --------
// One wave per 16-row tile. A = x[16 rows, k..k+3]; B = W[k..k+3] broadcast
// to all 16 output columns (every column of D equals the matvec result).
// f32 A 16x4 layout: lanes 0-15 -> M=lane, VGPR0=K0/VGPR1=K1;
//                    lanes 16-31 -> M=lane-16, VGPR0=K2/VGPR1=K3.
// D layout: VGPR r, lanes 0-15 -> M=r; lanes 16-31 -> M=r+8.
__global__ __launch_bounds__(256) void matvec_wmma(const float* __restrict__ x,
                                                   const float* __restrict__ W,
                                                   float* __restrict__ h) {
    const int lane = threadIdx.x & 31;
    const int wave = threadIdx.x >> 5;
    const int tile = blockIdx.x * 8 + wave;     // 8192 tiles total
    const int row  = lane & 15;
    const int kb   = (lane >> 4) << 1;          // 0 or 2
    const float* __restrict__ xr = x + (size_t)(tile * 16 + row) * FDIM;
    v8f acc = {};
    for (int k = 0; k < FDIM; k += 4) {
        v2f a, w;
        a[0] = xr[k + kb];     a[1] = xr[k + kb + 1];
        w[0] = W[k + kb];      w[1] = W[k + kb + 1];
        acc = __builtin_amdgcn_wmma_f32_16x16x4_f32(
            /*neg_a=*/false, a, /*neg_b=*/false, w,
            /*c_mod=*/(short)0, acc, /*reuse_a=*/false, /*reuse_b=*/false);
    }
    if ((lane & 15) == 0) {                     // lane 0: rows 0..7, lane 16: rows 8..15
        int base = tile * 16 + ((lane >> 4) << 3);
        #pragma unroll
        for (int r = 0; r < 8; ++r) h[base + r] = acc[r];
    }
}

// -------- 3. in-degree (targets = edge_index[1]) --------
__global__ __launch_bounds__(256) void deg_kernel(const int* __restrict__ ei,
                                                  float* __restrict__ deg) {
    int e = blockIdx.x * 256 + threadIdx.x;
    if (e < ETOT) atomicAdd(&deg[ei[ETOT + e]], 1.0f);
}

// -------- 4. dinv = rsqrt(deg); g = dinv * h --------
__global__ __launch_bounds__(256) void norm_kernel(const float* __restrict__ deg,
                                                   const float* __restrict__ h,
                                                   float* __restrict__ dinv,
                                                   float* __restrict__ g) {
    int i = blockIdx.x * 256 + threadIdx.x;
    if (i < NTOT) {
        float d = deg[i];                       // >= 1 always (self loop)
        float di = rsqrtf(d);
        dinv[i] = di;
        g[i] = di * h[i];
    }
}

// -------- 5. agg[dst] += g[src] --------
__global__ __launch_bounds__(256) void agg_kernel(const int* __restrict__ ei,
                                                  const float* __restrict__ g,
                                                  float* __restrict__ agg) {
    int e = blockIdx.x * 256 + threadIdx.x;
    if (e < ETOT) atomicAdd(&agg[ei[ETOT + e]], g[ei[e]]);
}

// -------- 6. score = dinv * (agg + g) + b   (self loop: dinv^2 * h = dinv*g) --------
__global__ __launch_bounds__(256) void score_kernel(const float* __restrict__ dinv,
                                                    const float* __restrict__ agg,
                                                    const float* __restrict__ g,
                                                    const float* __restrict__ bptr,
                                                    float* __restrict__ score) {
    int i = blockIdx.x * 256 + threadIdx.x;
    if (i < NTOT) score[i] = dinv[i] * (agg[i] + g[i]) + bptr[0];
}

// -------- 7. per-graph top-K by rank counting (stable descending == lax.top_k) --------
__global__ __launch_bounds__(128) void topk_kernel(const float* __restrict__ score,
                                                   int* __restrict__ perm_i,
                                                   int* __restrict__ new_idx,
                                                   float* __restrict__ out_batch,
                                                   float* __restrict__ out_perm) {
    __shared__ float s[NPER];
    const int b = blockIdx.x;
    const int i = threadIdx.x;
    const float v = score[b * NPER + i];
    s[i] = v;
    __syncthreads();
    int rank = 0;
    #pragma unroll 8
    for (int j = 0; j < NPER; ++j) {
        float sj = s[j];
        rank += (sj > v) || (sj == v && j < i);
    }
    if (rank < KEEP) {
        int gidx = b * KEEP + rank;             // new contiguous id
        int orig = b * NPER + i;
        perm_i[gidx]   = orig;
        new_idx[orig]  = gidx;
        out_batch[gidx] = (float)b;
        out_perm[gidx]  = (float)orig;
    }
}

// -------- 8. x_pool = x[perm] * tanh(score[perm]), float4 streams, 4 rows/block --------
__global__ __launch_bounds__(256) void gather_kernel(const float* __restrict__ x,
                                                     const float* __restrict__ score,
                                                     const int* __restrict__ perm_i,
                                                     float* __restrict__ out_x) {
    int row = blockIdx.x * 4 + (threadIdx.x >> 6);
    int c   = (threadIdx.x & 63) << 2;          // 64 lanes * float4 = 256 floats
    int orig = perm_i[row];
    float gate = tanhf(score[orig]);
    const v4f* src = (const v4f*)(x + (size_t)orig * FDIM + c);
    v4f val = *src;
    val *= gate;
    *(v4f*)(out_x + (size_t)row * FDIM + c) = val;
}

// -------- 9. edge remap + mask --------
__global__ __launch_bounds__(256) void edge_kernel(const int* __restrict__ ei,
                                                   const int* __restrict__ new_idx,
                                                   float* __restrict__ out_ei,
                                                   float* __restrict__ out_mask) {
    int e = blockIdx.x * 256 + threadIdx.x;
    if (e < ETOT) {
        int m0 = new_idx[ei[e]];
        int m1 = new_idx[ei[ETOT + e]];
        bool keep = (m0 >= 0) && (m1 >= 0);
        out_ei[e]        = (float)(keep ? m0 : -1);
        out_ei[ETOT + e] = (float)(keep ? m1 : -1);
        out_mask[e]      = keep ? 1.0f : 0.0f;
    }
}

extern "C" void kernel_launch(void* const* d_in, const int* in_sizes, int n_in,
                              void* d_out, int out_size, void* d_ws, size_t ws_size,
                              hipStream_t stream) {
    const float* x  = (const float*)d_in[0];
    const float* W  = (const float*)d_in[1];
    const float* bb = (const float*)d_in[2];
    const int*   ei = (const int*)d_in[3];
    // d_in[4] (batch) recomputed analytically

    // workspace carve-up
    char* ws = (char*)d_ws;
    float* h     = (float*)ws;                 ws += (size_t)NTOT * 4;
    float* deg   = (float*)ws;                 ws += (size_t)NTOT * 4;
    float* dinv  = (float*)ws;                 ws += (size_t)NTOT * 4;
    float* g     = (float*)ws;                 ws += (size_t)NTOT * 4;
    float* agg   = (float*)ws;                 ws += (size_t)NTOT * 4;
    float* score = (float*)ws;                 ws += (size_t)NTOT * 4;
    int*   perm_i  = (int*)ws;                 ws += (size_t)BK   * 4;
    int*   new_idx = (int*)ws;                 ws += (size_t)NTOT * 4;

    // output carve-up (reference return order, flattened)
    float* out_x     = (float*)d_out;                  // [BK, FDIM]
    float* out_ei    = out_x    + (size_t)BK * FDIM;   // [2, ETOT]
    float* out_mask  = out_ei   + (size_t)2 * ETOT;    // [ETOT]
    float* out_batch = out_mask + (size_t)ETOT;        // [BK]
    float* out_perm  = out_batch + (size_t)BK;         // [BK]

    init_kernel<<<(NTOT + 255) / 256, 256, 0, stream>>>(deg, agg, new_idx);
    matvec_wmma<<<NTOT / 128, 256, 0, stream>>>(x, W, h);          // 8 waves/block, 16 rows/wave
    deg_kernel<<<ETOT / 256, 256, 0, stream>>>(ei, deg);
    norm_kernel<<<(NTOT + 255) / 256, 256, 0, stream>>>(deg, h, dinv, g);
    agg_kernel<<<ETOT / 256, 256, 0, stream>>>(ei, g, agg);
    score_kernel<<<(NTOT + 255) / 256, 256, 0, stream>>>(dinv, agg, g, bb, score);
    topk_kernel<<<BGR, 128, 0, stream>>>(score, perm_i, new_idx, out_batch, out_perm);
    gather_kernel<<<BK / 4, 256, 0, stream>>>(x, score, perm_i, out_x);
    edge_kernel<<<ETOT / 256, 256, 0, stream>>>(ei, new_idx, out_ei, out_mask);
}